// HGCNEncoder_39573828666117
// MI455X (gfx1250) — compile-verified
//
#include <hip/hip_runtime.h>
#include <hip/hip_bf16.h>

// ---------------------------------------------------------------------------
// HGCN encoder for MI455X (gfx1250, wave32).
//   - fp32 GEMMs via V_WMMA_F32_16X16X4_F32 (one 16x16 tile per wave)
//   - sorted-edge run-compressed segment sums (register accumulate, atomic flush)
//   - node scatters via global_atomic_add_f32 (random targets)
// ---------------------------------------------------------------------------

#define N_NODES  50000
#define N_HEDGES 100000
#define NNZ_     1600000
#define D_IN_    128
#define H0_      256
#define H1_      128

typedef __attribute__((ext_vector_type(2))) float v2f;
typedef __attribute__((ext_vector_type(8))) float v8f;

// ---------------------------------------------------------------------------
// Degree / inverse-degree
// ---------------------------------------------------------------------------
__global__ void degree_kernel(const int* __restrict__ nidx,
                              const int* __restrict__ eidx,
                              float* __restrict__ deg,
                              float* __restrict__ bdeg, int nnz) {
    int i = blockIdx.x * blockDim.x + threadIdx.x;
    if (i < nnz) {
        atomicAdd(&deg[nidx[i]], 1.0f);
        atomicAdd(&bdeg[eidx[i]], 1.0f);
    }
}

__global__ void invert_kernel(float* __restrict__ p, int n) {
    int i = blockIdx.x * blockDim.x + threadIdx.x;
    if (i < n) {
        float v = p[i];
        p[i] = (v > 0.0f) ? (1.0f / v) : 0.0f;
    }
}

// ---------------------------------------------------------------------------
// fp32 WMMA GEMM: Y[M,N] = X[M,K] @ W[K,N]
// One wave computes one 16x16 output tile, stepping K by 4.
// A (16x4):  lanes 0-15 -> M=lane, K=0,1 ; lanes 16-31 -> M=lane-16, K=2,3
// B (4x16):  VGPR0/1 rows K={0,1} on lanes 0-15, K={2,3} on lanes 16-31
// C/D (16x16): VGPR r -> row r (lanes 0-15) / row r+8 (lanes 16-31), N=lane%16
// M % 16 == 0, N % 16 == 0, K % 4 == 0 (guaranteed by problem sizes).
// ---------------------------------------------------------------------------
__global__ void wmma_gemm_f32(const float* __restrict__ X,
                              const float* __restrict__ W,
                              float* __restrict__ Y,
                              int M, int N, int K) {
    int waves_per_block = blockDim.x >> 5;
    int wave = blockIdx.x * waves_per_block + (threadIdx.x >> 5);
    int tilesN = N >> 4;
    int tiles  = (M >> 4) * tilesN;
    if (wave >= tiles) return;              // wave-uniform: EXEC stays all-ones

    int tm   = wave / tilesN;
    int tn   = wave % tilesN;
    int lane = threadIdx.x & 31;
    int half = lane >> 4;                   // 0: K-lo pair, 1: K-hi pair
    int l16  = lane & 15;

    const float* xrow = X + (size_t)(tm * 16 + l16) * K + 2 * half;   // A fetch base
    const float* wcol = W + (size_t)(2 * half) * N + tn * 16 + l16;   // B fetch base

    v8f acc = {};
#pragma unroll 8
    for (int k = 0; k < K; k += 4) {
        v2f a = *(const v2f*)(xrow + k);            // K = k+2h, k+2h+1 (contig)
        v2f b;
        b.x = wcol[(size_t)k * N];                  // row k+2h
        b.y = wcol[(size_t)(k + 1) * N];            // row k+2h+1
        acc = __builtin_amdgcn_wmma_f32_16x16x4_f32(
                  false, a, false, b, (short)0, acc, false, false);
    }

    float* out = Y + (size_t)(tm * 16 + half * 8) * N + tn * 16 + l16;
#pragma unroll
    for (int r = 0; r < 8; ++r)
        out[(size_t)r * N] = acc[r];
}

// ---------------------------------------------------------------------------
// Edge segment sum: dst[e, :] += sum over nnz with edge_idx==e of src[node_idx, :]
// edge_idx is SORTED -> run-compress in registers, flush with atomics only at
// run boundaries. Indices staged in LDS once per block.
// blockDim = 256, each block covers NZB nnz entries, thread owns (sub-strip, quad).
// ---------------------------------------------------------------------------
#define NZB 128

__global__ void edge_segsum_kernel(const float* __restrict__ src,
                                   const int* __restrict__ nidx,
                                   const int* __restrict__ eidx,
                                   float* __restrict__ dst,
                                   int C, int nnz) {
    __shared__ int s_nid[NZB];
    __shared__ int s_eid[NZB];
    int base = blockIdx.x * NZB;
    int t = threadIdx.x;

    for (int i = t; i < NZB; i += blockDim.x) {
        int nz = base + i;
        s_nid[i] = (nz < nnz) ? nidx[nz] : 0;
        s_eid[i] = (nz < nnz) ? eidx[nz] : -1;
    }
    __syncthreads();

    int quads   = C >> 2;                 // float4 lanes per row (32 or 64)
    int q       = t % quads;              // channel quad (coalesced across threads)
    int s       = t / quads;              // sub-strip within block
    int nstrips = blockDim.x / quads;     // 4 (C=256) or 8 (C=128)
    int slen    = NZB / nstrips;

    int j0 = s * slen;
    int j1 = j0 + slen;

    float4 acc = {0.f, 0.f, 0.f, 0.f};
    int cur = -1;
    for (int j = j0; j < j1; ++j) {
        int nz = base + j;
        if (nz >= nnz) break;
        int e = s_eid[j];
        if (e != cur) {
            if (cur >= 0) {
                float* p = dst + (size_t)cur * C + 4 * q;
                atomicAdd(p + 0, acc.x); atomicAdd(p + 1, acc.y);
                atomicAdd(p + 2, acc.z); atomicAdd(p + 3, acc.w);
            }
            cur = e;
            acc = {0.f, 0.f, 0.f, 0.f};
        }
        const float4 v = *(const float4*)(src + (size_t)s_nid[j] * C + 4 * q);
        acc.x += v.x; acc.y += v.y; acc.z += v.z; acc.w += v.w;
    }
    if (cur >= 0) {
        float* p = dst + (size_t)cur * C + 4 * q;
        atomicAdd(p + 0, acc.x); atomicAdd(p + 1, acc.y);
        atomicAdd(p + 2, acc.z); atomicAdd(p + 3, acc.w);
    }
}

// ---------------------------------------------------------------------------
// Node scatter: dst[node_idx[i], :] += mE[edge_idx[i], :] * Binv[edge_idx[i]]
// Random node targets -> plain atomics; consecutive threads share nz (coalesced
// float4 gathers that hit L2 since the edge buffer fits in 192MB L2).
// ---------------------------------------------------------------------------
__global__ void node_scatter_kernel(const float* __restrict__ mE,
                                    const float* __restrict__ Binv,
                                    const int* __restrict__ nidx,
                                    const int* __restrict__ eidx,
                                    float* __restrict__ dst,
                                    int C, int nnz) {
    int quads = C >> 2;
    long long tid   = (long long)blockIdx.x * blockDim.x + threadIdx.x;
    long long total = (long long)nnz * quads;
    if (tid >= total) return;
    int q  = (int)(tid % quads);
    int nz = (int)(tid / quads);
    int e = eidx[nz];
    int n = nidx[nz];
    float sc = Binv[e];
    const float4 v = *(const float4*)(mE + (size_t)e * C + 4 * q);
    float* p = dst + (size_t)n * C + 4 * q;
    atomicAdd(p + 0, v.x * sc); atomicAdd(p + 1, v.y * sc);
    atomicAdd(p + 2, v.z * sc); atomicAdd(p + 3, v.w * sc);
}

// ---------------------------------------------------------------------------
// y[m, c] = y[m, c] * Dinv[m] + bias[c]   (+ optional leaky ReLU 0.01)
// ---------------------------------------------------------------------------
__global__ void finalize_kernel(float* __restrict__ y,
                                const float* __restrict__ Dinv,
                                const float* __restrict__ bias,
                                int C, int M, int leaky) {
    long long tid   = (long long)blockIdx.x * blockDim.x + threadIdx.x;
    long long total = (long long)M * C;
    if (tid >= total) return;
    int m = (int)(tid / C);
    int c = (int)(tid % C);
    float v = y[tid] * Dinv[m] + bias[c];
    if (leaky) v = (v > 0.f) ? v : 0.01f * v;
    y[tid] = v;
}

// ---------------------------------------------------------------------------
// Host-side helpers
// ---------------------------------------------------------------------------
static inline void launch_gemm(const float* X, const float* W, float* Y,
                               int M, int N, int K, hipStream_t s) {
    int tiles  = (M >> 4) * (N >> 4);
    int blocks = (tiles + 7) / 8;          // 8 waves (256 threads) per block
    hipLaunchKernelGGL(wmma_gemm_f32, dim3(blocks), dim3(256), 0, s, X, W, Y, M, N, K);
}

static inline void launch_edge_segsum(const float* src, const int* nidx, const int* eidx,
                                      float* dst, int C, hipStream_t s) {
    int blocks = (NNZ_ + NZB - 1) / NZB;
    hipLaunchKernelGGL(edge_segsum_kernel, dim3(blocks), dim3(256), 0, s,
                       src, nidx, eidx, dst, C, NNZ_);
}

static inline void launch_node_scatter(const float* mE, const float* Binv,
                                       const int* nidx, const int* eidx,
                                       float* dst, int C, hipStream_t s) {
    long long total  = (long long)NNZ_ * (C >> 2);
    int       blocks = (int)((total + 255) / 256);
    hipLaunchKernelGGL(node_scatter_kernel, dim3(blocks), dim3(256), 0, s,
                       mE, Binv, nidx, eidx, dst, C, NNZ_);
}

static inline void launch_finalize(float* y, const float* Dinv, const float* bias,
                                   int C, int M, int leaky, hipStream_t s) {
    long long total  = (long long)M * C;
    int       blocks = (int)((total + 255) / 256);
    hipLaunchKernelGGL(finalize_kernel, dim3(blocks), dim3(256), 0, s,
                       y, Dinv, bias, C, M, leaky);
}

extern "C" void kernel_launch(void* const* d_in, const int* in_sizes, int n_in,
                              void* d_out, int out_size, void* d_ws, size_t ws_size,
                              hipStream_t stream) {
    const float* x    = (const float*)d_in[0];
    const float* W0   = (const float*)d_in[1];
    const float* b0   = (const float*)d_in[2];
    const float* W1   = (const float*)d_in[3];
    const float* b1   = (const float*)d_in[4];
    const float* Wmu  = (const float*)d_in[5];
    const float* bmu  = (const float*)d_in[6];
    const float* Wlv  = (const float*)d_in[7];
    const float* blv  = (const float*)d_in[8];
    const int*   nidx = (const int*)d_in[9];
    const int*   eidx = (const int*)d_in[10];

    float* out = (float*)d_out;

    // Workspace carve-up (floats)
    float* Dinv = (float*)d_ws;                         //   50,000 (padded)
    float* Binv = Dinv + 50048;                         //  100,000 (padded)
    float* bufA = Binv + 100096;                        // 50000*256 GEMM out
    float* bufE = bufA + (size_t)N_NODES * H0_;         // 100000*256 edge acc
    float* bufB = bufE + (size_t)N_HEDGES * H0_;        // 50000*256 h / lv
    float* bufC = bufB + (size_t)N_NODES * H0_;         // 50000*128 h2
    float* bufD = bufC + (size_t)N_NODES * H1_;         // 50000*128 mu

    // ---- degrees ----
    hipMemsetAsync(Dinv, 0, (size_t)N_NODES  * sizeof(float), stream);
    hipMemsetAsync(Binv, 0, (size_t)N_HEDGES * sizeof(float), stream);
    hipLaunchKernelGGL(degree_kernel, dim3((NNZ_ + 255) / 256), dim3(256), 0, stream,
                       nidx, eidx, Dinv, Binv, NNZ_);
    hipLaunchKernelGGL(invert_kernel, dim3((N_NODES + 255) / 256), dim3(256), 0, stream,
                       Dinv, N_NODES);
    hipLaunchKernelGGL(invert_kernel, dim3((N_HEDGES + 255) / 256), dim3(256), 0, stream,
                       Binv, N_HEDGES);

    // ---- layer 0: h = leaky(hconv(x, W0, b0)) -> bufB [N,256] ----
    launch_gemm(x, W0, bufA, N_NODES, H0_, D_IN_, stream);
    hipMemsetAsync(bufE, 0, (size_t)N_HEDGES * H0_ * sizeof(float), stream);
    launch_edge_segsum(bufA, nidx, eidx, bufE, H0_, stream);
    hipMemsetAsync(bufB, 0, (size_t)N_NODES * H0_ * sizeof(float), stream);
    launch_node_scatter(bufE, Binv, nidx, eidx, bufB, H0_, stream);
    launch_finalize(bufB, Dinv, b0, H0_, N_NODES, 1, stream);

    // ---- layer 1: h2 = hconv(h, W1, b1) -> bufC [N,128] ----
    launch_gemm(bufB, W1, bufA, N_NODES, H1_, H0_, stream);
    hipMemsetAsync(bufE, 0, (size_t)N_HEDGES * H1_ * sizeof(float), stream);
    launch_edge_segsum(bufA, nidx, eidx, bufE, H1_, stream);
    hipMemsetAsync(bufC, 0, (size_t)N_NODES * H1_ * sizeof(float), stream);
    launch_node_scatter(bufE, Binv, nidx, eidx, bufC, H1_, stream);
    launch_finalize(bufC, Dinv, b1, H1_, N_NODES, 0, stream);

    // ---- mu = hconv(h2, Wmu, bmu) -> bufD [N,128] ----
    launch_gemm(bufC, Wmu, bufA, N_NODES, H1_, H1_, stream);
    hipMemsetAsync(bufE, 0, (size_t)N_HEDGES * H1_ * sizeof(float), stream);
    launch_edge_segsum(bufA, nidx, eidx, bufE, H1_, stream);
    hipMemsetAsync(bufD, 0, (size_t)N_NODES * H1_ * sizeof(float), stream);
    launch_node_scatter(bufE, Binv, nidx, eidx, bufD, H1_, stream);
    launch_finalize(bufD, Dinv, bmu, H1_, N_NODES, 0, stream);

    // ---- lv = hconv(h2, Wlv, blv) -> bufB (reused) [N,128] ----
    launch_gemm(bufC, Wlv, bufA, N_NODES, H1_, H1_, stream);
    hipMemsetAsync(bufE, 0, (size_t)N_HEDGES * H1_ * sizeof(float), stream);
    launch_edge_segsum(bufA, nidx, eidx, bufE, H1_, stream);
    hipMemsetAsync(bufB, 0, (size_t)N_NODES * H1_ * sizeof(float), stream);
    launch_node_scatter(bufE, Binv, nidx, eidx, bufB, H1_, stream);
    launch_finalize(bufB, Dinv, blv, H1_, N_NODES, 0, stream);

    // ---- pooling: mu_pool | lv_pool -> d_out ----
    hipMemsetAsync(out, 0, (size_t)2 * N_HEDGES * H1_ * sizeof(float), stream);
    launch_edge_segsum(bufD, nidx, eidx, out, H1_, stream);
    launch_edge_segsum(bufB, nidx, eidx, out + (size_t)N_HEDGES * H1_, H1_, stream);
}